// Encoder_54924041781853
// MI455X (gfx1250) — compile-verified
//
#include <hip/hip_runtime.h>
#include <hip/hip_bf16.h>

#define NNODES 100000
#define NEDGES 3200000
#define NNZ    3200000
#define KWAV   3
#define INC    256
#define OUTC   128
#define ATTH   64

typedef __attribute__((ext_vector_type(2))) float v2f;
typedef __attribute__((ext_vector_type(8))) float v8f;

#if __has_builtin(__builtin_amdgcn_global_load_async_to_lds_b32)
#define HAVE_ASYNC_LDS 1
typedef __attribute__((address_space(1))) int gas_int;   // global
typedef __attribute__((address_space(3))) int las_int;   // LDS
#else
#define HAVE_ASYNC_LDS 0
#endif

__device__ __forceinline__ void wait_async0() {
#if __has_builtin(__builtin_amdgcn_s_wait_asynccnt)
    __builtin_amdgcn_s_wait_asynccnt(0);
#else
    asm volatile("s_wait_asynccnt 0x0" ::: "memory");
#endif
}

// ---------------------------------------------------------------------------
// fp32 atomic add lowering to global_atomic_add_f32
// ---------------------------------------------------------------------------
__device__ __forceinline__ void fatomic_add(float* p, float v) {
    unsafeAtomicAdd(p, v);
}

// ---------------------------------------------------------------------------
// WMMA fp32 GEMM: C[M x NOUT] = A[M x K] @ B[K x NOUT] (+bias) (EPI=1: relu)
// One block = 16 output rows, NOUT/16 waves, each wave owns one 16x16 tile.
// K staged through LDS in 64-wide chunks via async global->LDS when available.
// Uses V_WMMA_F32_16X16X4_F32 (full fp32 precision).
// ---------------------------------------------------------------------------
template <int K, int NOUT, int EPI>
__global__ __launch_bounds__(2 * NOUT)
void gemm_wmma_f32(const float* __restrict__ A, const float* __restrict__ B,
                   const float* __restrict__ bias, float* __restrict__ C) {
    constexpr int KC = 64;              // K chunk staged in LDS
    constexpr int BD = 2 * NOUT;        // block dim (32 * NOUT/16)
    __shared__ float lA[16 * (KC + 1)];
    __shared__ float lB[KC * (NOUT + 1)];

    const int tid  = threadIdx.x;
    const int wave = tid >> 5;
    const int lane = tid & 31;
    const int row0 = blockIdx.x * 16;
    const int mrow = lane & 15;         // A row within tile (lanes 0-15 / 16-31)
    const int hi   = lane >> 4;         // lane-half selects K pair (ISA 16x4 layout)
    const int ncol = wave * 16 + (lane & 15);  // output column owned by this lane

    const float bv = bias ? bias[ncol] : 0.0f;
    v8f acc = {};

    for (int k0 = 0; k0 < K; k0 += KC) {
#if HAVE_ASYNC_LDS
        // async copy A tile (16 x KC) and B tile (KC x NOUT) straight into LDS
        for (int i = tid; i < 16 * KC; i += BD) {
            int r = i / KC, c = i - r * KC;
            __builtin_amdgcn_global_load_async_to_lds_b32(
                (gas_int*)(void*)(A + (size_t)(row0 + r) * K + k0 + c),
                (las_int*)(void*)&lA[r * (KC + 1) + c], 0, 0);
        }
        for (int i = tid; i < KC * NOUT; i += BD) {
            int r = i / NOUT, c = i - r * NOUT;
            __builtin_amdgcn_global_load_async_to_lds_b32(
                (gas_int*)(void*)(B + (size_t)(k0 + r) * NOUT + c),
                (las_int*)(void*)&lB[r * (NOUT + 1) + c], 0, 0);
        }
        wait_async0();
#else
        for (int i = tid; i < 16 * KC; i += BD) {
            int r = i / KC, c = i - r * KC;
            lA[r * (KC + 1) + c] = A[(size_t)(row0 + r) * K + k0 + c];
        }
        for (int i = tid; i < KC * NOUT; i += BD) {
            int r = i / NOUT, c = i - r * NOUT;
            lB[r * (NOUT + 1) + c] = B[(size_t)(k0 + r) * NOUT + c];
        }
#endif
        __syncthreads();

#pragma unroll
        for (int kk = 0; kk < KC; kk += 4) {
            const int ka = kk + 2 * hi;
            v2f a, b;
            a.x = lA[mrow * (KC + 1) + ka];
            a.y = lA[mrow * (KC + 1) + ka + 1];
            b.x = lB[ka * (NOUT + 1) + ncol];
            b.y = lB[(ka + 1) * (NOUT + 1) + ncol];
            acc = __builtin_amdgcn_wmma_f32_16x16x4_f32(
                /*neg_a=*/false, a, /*neg_b=*/false, b,
                /*c_mod=*/(short)0, acc, /*reuse_a=*/false, /*reuse_b=*/false);
        }
        __syncthreads();
    }

    // C/D layout: vgpr r, lane l -> M = r + 8*(l/16), N = l%16 (+ wave*16)
#pragma unroll
    for (int r = 0; r < 8; ++r) {
        const int m = row0 + r + 8 * hi;
        float c = acc[r] + bv;
        if (EPI == 1) c = fmaxf(c, 0.0f);
        C[(size_t)m * NOUT + ncol] = c;
    }
}

// ---------------------------------------------------------------------------
// Row-wise L2 normalize * 1.8 (128 threads per row)
// ---------------------------------------------------------------------------
__global__ void l2norm_rows(float* __restrict__ M) {
    __shared__ float red[4];
    const int n = blockIdx.x, c = threadIdx.x;
    float v = M[(size_t)n * OUTC + c];
    float s = v * v;
#pragma unroll
    for (int o = 16; o; o >>= 1) s += __shfl_xor(s, o, 32);
    if ((c & 31) == 0) red[c >> 5] = s;
    __syncthreads();
    const float tot = red[0] + red[1] + red[2] + red[3];
    const float scale = 1.8f / fmaxf(sqrtf(tot), 1e-12f);
    M[(size_t)n * OUTC + c] = v * scale;
}

// ---------------------------------------------------------------------------
// Degree count (integer atomics -> deterministic), then dinv = rsqrt(deg+1)
// ---------------------------------------------------------------------------
__global__ void deg_count(const int* __restrict__ cols, int* __restrict__ degi, int nE) {
    int i = blockIdx.x * blockDim.x + threadIdx.x;
    if (i < nE) atomicAdd(&degi[cols[i]], 1);
}

__global__ void make_dinv(const int* __restrict__ degi, float* __restrict__ dinv, int n) {
    int i = blockIdx.x * blockDim.x + threadIdx.x;
    if (i < n) dinv[i] = rsqrtf((float)degi[i] + 1.0f);  // +1 self-loop
}

// ---------------------------------------------------------------------------
// APPNP edge scatter: out[row] += dinv[row]*dinv[col]*h[col]; 1 wave / edge.
// Lane owns 4 consecutive columns -> one b128 gather per lane (wave = 512B).
// ---------------------------------------------------------------------------
__global__ void appnp_edges(const int* __restrict__ rows, const int* __restrict__ cols,
                            const float* __restrict__ dinv, const float* __restrict__ h,
                            float* __restrict__ out, int nE) {
    const int wid  = (blockIdx.x * blockDim.x + threadIdx.x) >> 5;
    const int lane = threadIdx.x & 31;
    if (wid >= nE) return;
    const int r = rows[wid], c = cols[wid];
    const float nrm = dinv[r] * dinv[c];
    const float4 s = *(const float4*)(h + (size_t)c * OUTC + lane * 4);
    float* dst = out + (size_t)r * OUTC + lane * 4;
    fatomic_add(dst + 0, nrm * s.x);
    fatomic_add(dst + 1, nrm * s.y);
    fatomic_add(dst + 2, nrm * s.z);
    fatomic_add(dst + 3, nrm * s.w);
}

__global__ void appnp_selfloop(const float4* __restrict__ h, const float* __restrict__ dinv,
                               float4* __restrict__ out, int total4) {
    int i = blockIdx.x * blockDim.x + threadIdx.x;
    if (i >= total4) return;
    const float di = dinv[i >> 5];        // 32 float4 per 128-float row
    const float w = di * di;
    float4 hv = h[i];
    float4 ov = out[i];
    ov.x += w * hv.x; ov.y += w * hv.y; ov.z += w * hv.z; ov.w += w * hv.w;
    out[i] = ov;
}

// ---------------------------------------------------------------------------
// SpMM scatter: out[row*outStride + :] += vals[e] * in[col*inStride + :]
// ---------------------------------------------------------------------------
__global__ void spmm_atomic(const int* __restrict__ rows, const int* __restrict__ cols,
                            const float* __restrict__ vals, const float* __restrict__ in,
                            int inStride, float* __restrict__ out, int outStride, int nnz) {
    const int wid  = (blockIdx.x * blockDim.x + threadIdx.x) >> 5;
    const int lane = threadIdx.x & 31;
    if (wid >= nnz) return;
    const int r = rows[wid], c = cols[wid];
    const float v = vals[wid];
    const float4 s = *(const float4*)(in + (size_t)c * inStride + lane * 4);
    float* dst = out + (size_t)r * outStride + lane * 4;
    fatomic_add(dst + 0, v * s.x);
    fatomic_add(dst + 1, v * s.y);
    fatomic_add(dst + 2, v * s.z);
    fatomic_add(dst + 3, v * s.w);
}

__global__ void diag_mul(float4* __restrict__ t, const float* __restrict__ dw, int total4) {
    int i = blockIdx.x * blockDim.x + threadIdx.x;
    if (i >= total4) return;
    const float d = dw[i >> 5];
    float4 v = t[i];
    v.x *= d; v.y *= d; v.z *= d; v.w *= d;
    t[i] = v;
}

// ---------------------------------------------------------------------------
// logits[r] = H[r,:64] . w2   (H already relu'd by GEMM epilogue); 1 wave/row
// ---------------------------------------------------------------------------
__global__ void att_logits(const float* __restrict__ H, const float* __restrict__ w2,
                           float* __restrict__ logits, int R) {
    const int wid  = (blockIdx.x * blockDim.x + threadIdx.x) >> 5;
    const int lane = threadIdx.x & 31;
    if (wid >= R) return;
    const float* h = H + (size_t)wid * ATTH;
    float s = h[lane] * w2[lane] + h[lane + 32] * w2[lane + 32];
#pragma unroll
    for (int o = 16; o; o >>= 1) s += __shfl_xor(s, o, 32);
    if (lane == 0) logits[wid] = s;
}

// ---------------------------------------------------------------------------
// Softmax over K=3 wavelets + weighted sum: out[n,:] = sum_k beta_k z[n,k,:]
// ---------------------------------------------------------------------------
__global__ void attpool(const float* __restrict__ z, const float* __restrict__ logits,
                        float* __restrict__ out) {
    const int n = blockIdx.x, d = threadIdx.x;
    const float l0 = logits[n * 3 + 0], l1 = logits[n * 3 + 1], l2 = logits[n * 3 + 2];
    const float m  = fmaxf(l0, fmaxf(l1, l2));
    const float e0 = expf(l0 - m), e1 = expf(l1 - m), e2 = expf(l2 - m);
    const float inv = 1.0f / (e0 + e1 + e2);
    const float* zn = z + (size_t)n * (KWAV * OUTC);
    out[(size_t)n * OUTC + d] =
        (e0 * zn[d] + e1 * zn[OUTC + d] + e2 * zn[2 * OUTC + d]) * inv;
}

// ---------------------------------------------------------------------------
extern "C" void kernel_launch(void* const* d_in, const int* in_sizes, int n_in,
                              void* d_out, int out_size, void* d_ws, size_t ws_size,
                              hipStream_t stream) {
    const float* x            = (const float*)d_in[0];
    const int*   edge         = (const int*)d_in[1];      // [2, E]
    const int*   phi_idx      = (const int*)d_in[2];      // [K, 2, NNZ]
    const float* phi_vals     = (const float*)d_in[3];    // [K, NNZ]
    const int*   phi_inv_idx  = (const int*)d_in[4];
    const float* phi_inv_vals = (const float*)d_in[5];
    const float* lin_w        = (const float*)d_in[6];    // [256,128]
    const float* lin_b        = (const float*)d_in[7];
    const float* gw_w         = (const float*)d_in[8];
    const float* diag_w       = (const float*)d_in[9];
    const float* att_w1       = (const float*)d_in[10];   // [128,64]
    const float* att_b1       = (const float*)d_in[11];
    const float* att_w2       = (const float*)d_in[12];   // [64,1]

    float* out_mean   = (float*)d_out;                    // [N,128]
    float* out_logstd = (float*)d_out + (size_t)NNODES * OUTC;

    // workspace layout (floats)
    float* ws     = (float*)d_ws;
    float* buf0   = ws;                                   // mean_pre  [N,128]
    float* buf1   = buf0 + (size_t)NNODES * OUTC;         // xw        [N,128]
    float* tbuf   = buf1 + (size_t)NNODES * OUTC;         // spmm tmp  [N,128]
    float* zbuf   = tbuf + (size_t)NNODES * OUTC;         // z         [N,3,128]
    float* logits = zbuf + (size_t)NNODES * KWAV * OUTC;  // [3N]
    float* dinv   = logits + (size_t)3 * NNODES;          // [N]
    int*   degi   = (int*)(dinv + NNODES);                // [N]
    float* hbuf   = buf0;  // attention hidden [3N,64] aliases buf0/buf1 (free by then)

    const int nvec  = NNODES * OUTC;
    const int nvec4 = nvec / 4;

    // ---- mean branch: linear + bias -> L2 normalize*1.8 -> APPNP(K=1,a=0)
    gemm_wmma_f32<INC, OUTC, 0><<<NNODES / 16, 256, 0, stream>>>(x, lin_w, lin_b, buf0);
    l2norm_rows<<<NNODES, OUTC, 0, stream>>>(buf0);

    (void)hipMemsetAsync(degi, 0, sizeof(int) * NNODES, stream);
    deg_count<<<(NEDGES + 255) / 256, 256, 0, stream>>>(edge + NEDGES, degi, NEDGES);
    make_dinv<<<(NNODES + 255) / 256, 256, 0, stream>>>(degi, dinv, NNODES);

    (void)hipMemsetAsync(out_mean, 0, sizeof(float) * (size_t)nvec, stream);
    appnp_edges<<<(NEDGES * 32 + 255) / 256, 256, 0, stream>>>(
        edge, edge + NEDGES, dinv, buf0, out_mean, NEDGES);
    appnp_selfloop<<<(nvec4 + 255) / 256, 256, 0, stream>>>(
        (const float4*)buf0, dinv, (float4*)out_mean, nvec4);

    // ---- logstd branch: x@gw_w -> K wavelet (spmm, diag, spmm)
    gemm_wmma_f32<INC, OUTC, 0><<<NNODES / 16, 256, 0, stream>>>(x, gw_w, nullptr, buf1);

    (void)hipMemsetAsync(zbuf, 0, sizeof(float) * (size_t)NNODES * KWAV * OUTC, stream);
    for (int k = 0; k < KWAV; ++k) {
        const int* pi_rows = phi_inv_idx + (size_t)k * 2 * NNZ;
        const int* pi_cols = pi_rows + NNZ;
        const int* p_rows  = phi_idx + (size_t)k * 2 * NNZ;
        const int* p_cols  = p_rows + NNZ;

        (void)hipMemsetAsync(tbuf, 0, sizeof(float) * (size_t)nvec, stream);
        spmm_atomic<<<(NNZ * 32 + 255) / 256, 256, 0, stream>>>(
            pi_rows, pi_cols, phi_inv_vals + (size_t)k * NNZ, buf1, OUTC, tbuf, OUTC, NNZ);
        diag_mul<<<(nvec4 + 255) / 256, 256, 0, stream>>>((float4*)tbuf, diag_w, nvec4);
        spmm_atomic<<<(NNZ * 32 + 255) / 256, 256, 0, stream>>>(
            p_rows, p_cols, phi_vals + (size_t)k * NNZ, tbuf, OUTC,
            zbuf + (size_t)k * OUTC, KWAV * OUTC, NNZ);
    }

    // ---- attention pooling over wavelet axis
    // H = relu(z@att_w1 + b1) as [3N,64] via WMMA, then dot with att_w2
    gemm_wmma_f32<OUTC, ATTH, 1><<<(3 * NNODES) / 16, 128, 0, stream>>>(
        zbuf, att_w1, att_b1, hbuf);
    att_logits<<<(3 * NNODES * 32 + 255) / 256, 256, 0, stream>>>(
        hbuf, att_w2, logits, 3 * NNODES);
    attpool<<<NNODES, OUTC, 0, stream>>>(zbuf, logits, out_logstd);
}